// NodeGNNwithAttention_9259949490533
// MI455X (gfx1250) — compile-verified
//
#include <hip/hip_runtime.h>
#include <hip/hip_bf16.h>

// ---------------------------------------------------------------------------
// GAT (2-layer) for MI455X / gfx1250, wave32.
//  - hid = h @ w        : fp32 WMMA (v_wmma_f32_16x16x4_f32), LDS-staged tiles,
//                         64-row block tile -> 4 WMMAs share each B fragment.
//                         Single manually-partitioned LDS block pins the A tile
//                         at LDS offset 0 so A-fragment loads use the 16-bit DS
//                         immediate offset (no per-iteration address VALU).
//  - s_in/s_out         : wave-per-row dot products, shfl_xor reduction
//  - softmax+gather SpMM: wave-per-node; hid (51.2MB) is L2-resident (192MB L2),
//                         so the 1.6GB of neighbor gathers run at L2 bandwidth.
// ---------------------------------------------------------------------------

typedef __attribute__((ext_vector_type(2))) float v2f;
typedef __attribute__((ext_vector_type(8))) float v8f;

#define GAT_ALPHA 0.2f
#define LDA 132          // padded LDS stride for A tile (132 % 64 != 0 -> no bank conflict)
#define BROWS 64         // rows per block
#define AS_FLOATS (BROWS * LDA)        // 8448 floats  = 33792 B  (A tile @ offset 0)
#define WS_FLOATS (128 * 128)          // 16384 floats = 65536 B  (W @ offset 33792)

// -------------------------- GEMM: hid = A @ W ------------------------------
// Block = 64 rows x 128 cols; 8 waves. Wave w owns column tile col0 = w*16 and
// computes 4 stacked 16x16 row tiles, sharing one B fragment per k-step.
__global__ __launch_bounds__(256)
void gat_gemm_kernel(const float* __restrict__ A,   // [N,128]
                     const float* __restrict__ W,   // [128,128]
                     float* __restrict__ Hid,       // [N,128]
                     int N)
{
    // One LDS object -> layout is ours: As at byte 0, Ws at byte 33792.
    __shared__ float smem[AS_FLOATS + WS_FLOATS];   // 97 KB
    float* __restrict__ As = smem;
    float* __restrict__ Ws = smem + AS_FLOATS;

    const int tid  = threadIdx.x;
    const int wave = tid >> 5;          // 0..7
    const int lane = tid & 31;
    const int row0 = blockIdx.x * BROWS;

    // Stage W (16384 floats) cooperatively: 256 threads x float4 x 16 sweeps.
    for (int i = tid * 4; i < WS_FLOATS; i += 256 * 4) {
        *(float4*)&Ws[i] = *(const float4*)&W[i];
    }
    // Stage A tile: 64 rows x 128 cols (row-clamped for the tail block).
    for (int i = tid * 4; i < BROWS * 128; i += 256 * 4) {
        int r  = i >> 7;        // row within tile
        int c  = i & 127;       // col
        int gr = row0 + r;
        if (gr >= N) gr = N - 1;
        *(float4*)&As[r * LDA + c] = *(const float4*)&A[(size_t)gr * 128 + c];
    }
    __syncthreads();

    // WMMA fragment indexing (ISA 7.12.2, 32-bit 16x16x4 layouts):
    //   A: lane m = lane%16 holds A[m][kk + 2*half + v] in a[v]
    //   B: lane n = lane%16 holds W[kk + 2*half + v][col0+n] in b[v]
    //   D: c[r] = out[r + 8*half][n] of the 16x16 tile
    const int m    = lane & 15;
    const int n    = lane & 15;
    const int half = lane >> 4;
    const int col0 = wave * 16;

    v8f c0 = {}, c1 = {}, c2 = {}, c3 = {};
#pragma unroll
    for (int kk = 0; kk < 128; kk += 4) {
        v2f b;
        b[0] = Ws[(kk + 2 * half + 0) * 128 + col0 + n];
        b[1] = Ws[(kk + 2 * half + 1) * 128 + col0 + n];
        v2f a0, a1, a2, a3;
        const int ka = kk + 2 * half;
        a0[0] = As[(m +  0) * LDA + ka];  a0[1] = As[(m +  0) * LDA + ka + 1];
        a1[0] = As[(m + 16) * LDA + ka];  a1[1] = As[(m + 16) * LDA + ka + 1];
        a2[0] = As[(m + 32) * LDA + ka];  a2[1] = As[(m + 32) * LDA + ka + 1];
        a3[0] = As[(m + 48) * LDA + ka];  a3[1] = As[(m + 48) * LDA + ka + 1];
        // D = A x B + C   (8 args: neg_a, A, neg_b, B, c_mod, C, reuse_a, reuse_b)
        c0 = __builtin_amdgcn_wmma_f32_16x16x4_f32(false, a0, false, b, (short)0, c0, false, false);
        c1 = __builtin_amdgcn_wmma_f32_16x16x4_f32(false, a1, false, b, (short)0, c1, false, false);
        c2 = __builtin_amdgcn_wmma_f32_16x16x4_f32(false, a2, false, b, (short)0, c2, false, false);
        c3 = __builtin_amdgcn_wmma_f32_16x16x4_f32(false, a3, false, b, (short)0, c3, false, false);
    }

#pragma unroll
    for (int r = 0; r < 8; ++r) {
        int rr = row0 + r + 8 * half;
        if (rr      < N) Hid[(size_t)(rr     ) * 128 + col0 + n] = c0[r];
        if (rr + 16 < N) Hid[(size_t)(rr + 16) * 128 + col0 + n] = c1[r];
        if (rr + 32 < N) Hid[(size_t)(rr + 32) * 128 + col0 + n] = c2[r];
        if (rr + 48 < N) Hid[(size_t)(rr + 48) * 128 + col0 + n] = c3[r];
    }
}

// ------------------- scores: s_in = hid@a_in, s_out = hid@a_out ------------
// One wave per row; lane owns 4 contiguous channels.
__global__ __launch_bounds__(256)
void gat_scores_kernel(const float* __restrict__ Hid,
                       const float* __restrict__ a_in,
                       const float* __restrict__ a_out,
                       float* __restrict__ s_in,
                       float* __restrict__ s_out,
                       int N)
{
    const int wave = threadIdx.x >> 5;
    const int lane = threadIdx.x & 31;
    const int row  = blockIdx.x * 8 + wave;
    if (row >= N) return;

    float4 h  = *(const float4*)&Hid[(size_t)row * 128 + lane * 4];
    float4 ai = *(const float4*)&a_in[lane * 4];
    float4 ao = *(const float4*)&a_out[lane * 4];

    float si = h.x * ai.x + h.y * ai.y + h.z * ai.z + h.w * ai.w;
    float so = h.x * ao.x + h.y * ao.y + h.z * ao.z + h.w * ao.w;
#pragma unroll
    for (int off = 16; off > 0; off >>= 1) {
        si += __shfl_xor(si, off, 32);
        so += __shfl_xor(so, off, 32);
    }
    if (lane == 0) {
        s_in[row]  = si;
        s_out[row] = so;
    }
}

// --------------- attention softmax + weighted neighbor gather --------------
// One wave per node. Lanes 0-15 own the 16 edges; all 32 lanes then do the
// float4-per-lane gather-accumulate over the 16 neighbor rows.
__global__ __launch_bounds__(256)
void gat_aggregate_kernel(const float* __restrict__ Hid,   // [N,128] (L2 resident)
                          const float* __restrict__ s_in,  // [N]
                          const float* __restrict__ s_out, // [N]
                          const int*   __restrict__ dst,   // [N,16]
                          const float* __restrict__ adj,   // [N,16]
                          const float* __restrict__ bias,  // [128]
                          float* __restrict__ Out,         // [N,128]
                          int N)
{
    const int wave = threadIdx.x >> 5;
    const int lane = threadIdx.x & 31;
    const int node = blockIdx.x * 8 + wave;
    if (node >= N) return;

    int   idx = 0;
    float wgt = 0.0f;
    float e   = -1e30f;
    if (lane < 16) {
        idx = dst[(size_t)node * 16 + lane];
        wgt = adj[(size_t)node * 16 + lane];
        float x = s_in[node] + s_out[idx];
        e = (x > 0.0f) ? x : GAT_ALPHA * x;   // leaky_relu
    }

    // softmax over the 16 edge lanes (width-16 butterfly; lanes 16-31 inert)
    float mx = e;
#pragma unroll
    for (int off = 8; off > 0; off >>= 1) mx = fmaxf(mx, __shfl_xor(mx, off, 16));
    float p = __expf(e - mx);
    float s = p;
#pragma unroll
    for (int off = 8; off > 0; off >>= 1) s += __shfl_xor(s, off, 16);
    float w = wgt * (p / s);

    // weighted gather: lane owns channels [lane*4, lane*4+4)
    float4 acc = {0.0f, 0.0f, 0.0f, 0.0f};
#pragma unroll
    for (int k = 0; k < 16; ++k) {
        float wk = __shfl(w,   k, 32);
        int   ik = __shfl(idx, k, 32);
        float4 hv = *(const float4*)&Hid[(size_t)ik * 128 + lane * 4];
        acc.x += wk * hv.x;
        acc.y += wk * hv.y;
        acc.z += wk * hv.z;
        acc.w += wk * hv.w;
    }
    float4 b = *(const float4*)&bias[lane * 4];
    acc.x = fmaxf(acc.x + b.x, 0.0f);
    acc.y = fmaxf(acc.y + b.y, 0.0f);
    acc.z = fmaxf(acc.z + b.z, 0.0f);
    acc.w = fmaxf(acc.w + b.w, 0.0f);
    *(float4*)&Out[(size_t)node * 128 + lane * 4] = acc;
}

// ---------------------------------------------------------------------------
extern "C" void kernel_launch(void* const* d_in, const int* in_sizes, int n_in,
                              void* d_out, int out_size, void* d_ws, size_t ws_size,
                              hipStream_t stream)
{
    const float* x      = (const float*)d_in[0];
    const int*   dst    = (const int*)  d_in[1];
    const float* adj    = (const float*)d_in[2];
    const float* w1     = (const float*)d_in[3];
    const float* a_in1  = (const float*)d_in[4];
    const float* a_out1 = (const float*)d_in[5];
    const float* b1     = (const float*)d_in[6];
    const float* w2     = (const float*)d_in[7];
    const float* a_in2  = (const float*)d_in[8];
    const float* a_out2 = (const float*)d_in[9];
    const float* b2     = (const float*)d_in[10];
    float*       out    = (float*)d_out;

    const int N = in_sizes[0] / 128;   // 100000

    // workspace layout (fp32): hid[N*128] | h1[N*128] | s_in[N] | s_out[N]
    float* ws   = (float*)d_ws;
    float* hid  = ws;
    float* h1   = hid  + (size_t)N * 128;
    float* sin_ = h1   + (size_t)N * 128;
    float* sout = sin_ + (size_t)N;

    dim3 blk(256);
    const int gemm_blocks = (N + BROWS - 1) / BROWS;
    const int row_blocks  = (N + 7) / 8;

    // ---- layer 1 ----
    gat_gemm_kernel     <<<gemm_blocks, blk, 0, stream>>>(x,  w1, hid, N);
    gat_scores_kernel   <<<row_blocks,  blk, 0, stream>>>(hid, a_in1, a_out1, sin_, sout, N);
    gat_aggregate_kernel<<<row_blocks,  blk, 0, stream>>>(hid, sin_, sout, dst, adj, b1, h1, N);

    // ---- layer 2 ----
    gat_gemm_kernel     <<<gemm_blocks, blk, 0, stream>>>(h1, w2, hid, N);
    gat_scores_kernel   <<<row_blocks,  blk, 0, stream>>>(hid, a_in2, a_out2, sin_, sout, N);
    gat_aggregate_kernel<<<row_blocks,  blk, 0, stream>>>(hid, sin_, sout, dst, adj, b2, out, N);
}